// ParticlesNetwork_2422361555701
// MI455X (gfx1250) — compile-verified
//
#include <hip/hip_runtime.h>
#include <stdint.h>

// Systematic particle resampling for MI455X (gfx1250).
//   N = 4,194,304 particles, D = 16 f32 state dims.
// Pipeline:
//   K1: per-block partial sums of weights        (1024 blocks x 4096 elems)
//   K2: exclusive scan of the 1024 block sums    (1 block)
//   K3: full inclusive cumsum -> d_ws            (1024 blocks)
//   K4: block-narrowed binary search per output row + 64B row copy via
//       CDNA5 async global->LDS->global data movers (ASYNCcnt path),
//       with NT temporal hints so the streams don't evict cums from L2.

#define N_PART 4194304
#define DDIM   16
#define BLOCK  256
#define ITEMS  16
#define CHUNK  (BLOCK * ITEMS)   // 4096 elems per block
#define NBLK   (N_PART / CHUNK)  // 1024 scan blocks

// ---------------------------------------------------------------- K1: block partial sums
__global__ __launch_bounds__(BLOCK) void k_partial_sums(const float* __restrict__ probs,
                                                        float* __restrict__ bsum) {
  __shared__ float red[BLOCK];
  const int t = threadIdx.x;
  const size_t base = (size_t)blockIdx.x * CHUNK + (size_t)t * ITEMS;
  const float4* p4 = (const float4*)(probs + base);
  float s = 0.0f;
#pragma unroll
  for (int j = 0; j < ITEMS / 4; ++j) {
    float4 v = p4[j];
    s += (v.x + v.y) + (v.z + v.w);
  }
  red[t] = s;
  __syncthreads();
  for (int off = BLOCK / 2; off > 0; off >>= 1) {
    if (t < off) red[t] += red[t + off];
    __syncthreads();
  }
  if (t == 0) bsum[blockIdx.x] = red[0];
}

// ---------------------------------------------------------------- K2: scan of block sums
__global__ __launch_bounds__(NBLK) void k_scan_block_sums(const float* __restrict__ bsum,
                                                          float* __restrict__ boff) {
  __shared__ float s[NBLK];
  const int t = threadIdx.x;
  s[t] = bsum[t];
  __syncthreads();
  // Hillis-Steele inclusive scan over 1024 values.
  for (int off = 1; off < NBLK; off <<= 1) {
    float v = (t >= off) ? s[t - off] : 0.0f;
    __syncthreads();
    s[t] += v;
    __syncthreads();
  }
  boff[t] = (t == 0) ? 0.0f : s[t - 1];  // exclusive prefix
}

// ---------------------------------------------------------------- K3: full inclusive cumsum
__global__ __launch_bounds__(BLOCK) void k_write_cumsum(const float* __restrict__ probs,
                                                        const float* __restrict__ boff,
                                                        float* __restrict__ cums) {
  __shared__ float tsum[BLOCK];
  const int t = threadIdx.x;
  const size_t base = (size_t)blockIdx.x * CHUNK + (size_t)t * ITEMS;
  const float4* p4 = (const float4*)(probs + base);
  float loc[ITEMS];
  float run = 0.0f;
#pragma unroll
  for (int j = 0; j < ITEMS / 4; ++j) {
    float4 v = p4[j];
    run += v.x; loc[4 * j + 0] = run;
    run += v.y; loc[4 * j + 1] = run;
    run += v.z; loc[4 * j + 2] = run;
    run += v.w; loc[4 * j + 3] = run;
  }
  tsum[t] = run;
  __syncthreads();
  for (int off = 1; off < BLOCK; off <<= 1) {
    float v = (t >= off) ? tsum[t - off] : 0.0f;
    __syncthreads();
    tsum[t] += v;
    __syncthreads();
  }
  const float add = boff[blockIdx.x] + ((t == 0) ? 0.0f : tsum[t - 1]);
  float4* c4 = (float4*)(cums + base);
#pragma unroll
  for (int j = 0; j < ITEMS / 4; ++j) {
    float4 v;
    v.x = loc[4 * j + 0] + add;
    v.y = loc[4 * j + 1] + add;
    v.z = loc[4 * j + 2] + add;
    v.w = loc[4 * j + 3] + add;
    c4[j] = v;
  }
}

// Left searchsorted over [lo, hi): smallest i in [lo, hi] with cums[i] >= o.
__device__ __forceinline__ int lower_bound_range(const float* __restrict__ cums,
                                                 float o, int lo, int hi) {
  while (lo < hi) {
    const int mid = (lo + hi) >> 1;
    if (cums[mid] < o) lo = mid + 1; else hi = mid;
  }
  return lo;
}

// ---------------------------------------------------------------- K4: search + async row copy
__global__ __launch_bounds__(BLOCK) void k_resample_gather(const float* __restrict__ cums,
                                                           const float* __restrict__ particles,
                                                           const float* __restrict__ rnd,
                                                           float* __restrict__ out) {
  __shared__ float stage[BLOCK * DDIM];  // 16 KB: one 64B row slot per lane
  __shared__ int bounds[2];
  const int t = threadIdx.x;
  const int k = blockIdx.x * BLOCK + t;

  const float step = 1.0f / (float)N_PART;
  const float r0 = rnd[0];
  const float o = r0 * step + (float)k * step;

  // Lanes 0/1: full-range search for the block's first/last offset, computed
  // with the IDENTICAL fp expression as the per-thread offsets above.
  if (t < 2) {
    const int kb = blockIdx.x * BLOCK + (t == 0 ? 0 : (BLOCK - 1));
    const float ob = r0 * step + (float)kb * step;
    bounds[t] = lower_bound_range(cums, ob, 0, N_PART);
  }
  __syncthreads();

  // Monotone offsets => every answer lies in [bounds[0], bounds[1]].
  // Clamp hi so no probe ever reads cums[N].
  const int lo_blk = bounds[0];
  int hi_blk = bounds[1] + 1;
  if (hi_blk > N_PART) hi_blk = N_PART;
  const int found = lower_bound_range(cums, o, lo_blk, hi_blk);
  const int idx = (found < N_PART - 1) ? found : (N_PART - 1);

  // 64B row copy: global -> LDS -> global via CDNA5 async data movers.
  // NT hints keep the 512 MB of streaming traffic from evicting cums (16 MB)
  // out of the 192 MB L2.
  const uint32_t lds  = (uint32_t)(size_t)(&stage[t * DDIM]);
  const uint64_t gsrc = (uint64_t)(const void*)(particles + (size_t)idx * DDIM);
  const uint64_t gdst = (uint64_t)(void*)(out + (size_t)k * DDIM);

#pragma unroll
  for (int j = 0; j < 4; ++j) {
    const uint32_t l = lds + 16u * j;
    const uint64_t a = gsrc + 16u * j;
    asm volatile("global_load_async_to_lds_b128 %0, %1, off th:TH_LOAD_NT"
                 :: "v"(l), "v"(a) : "memory");
  }
  // Wait for LDS writes of the async loads before consuming that LDS as store source.
  asm volatile("s_wait_asynccnt 0x0" ::: "memory");
#pragma unroll
  for (int j = 0; j < 4; ++j) {
    const uint64_t a = gdst + 16u * j;
    const uint32_t l = lds + 16u * j;
    asm volatile("global_store_async_from_lds_b128 %0, %1, off th:TH_STORE_NT"
                 :: "v"(a), "v"(l) : "memory");
  }
  asm volatile("s_wait_asynccnt 0x0" ::: "memory");
}

// ---------------------------------------------------------------- launch
extern "C" void kernel_launch(void* const* d_in, const int* in_sizes, int n_in,
                              void* d_out, int out_size, void* d_ws, size_t ws_size,
                              hipStream_t stream) {
  (void)in_sizes; (void)n_in; (void)out_size; (void)ws_size;

  const float* particles = (const float*)d_in[0];  // [N, 16] f32
  const float* probs     = (const float*)d_in[1];  // [N]     f32 (normalized)
  const float* rnd       = (const float*)d_in[2];  // [1]     f32
  float* out             = (float*)d_out;          // [N, 16] f32

  float* cums = (float*)d_ws;        // N floats  (16 MB)
  float* bsum = cums + N_PART;       // NBLK floats
  float* boff = bsum + NBLK;         // NBLK floats

  k_partial_sums   <<<NBLK,          BLOCK, 0, stream>>>(probs, bsum);
  k_scan_block_sums<<<1,             NBLK,  0, stream>>>(bsum, boff);
  k_write_cumsum   <<<NBLK,          BLOCK, 0, stream>>>(probs, boff, cums);
  k_resample_gather<<<N_PART / BLOCK, BLOCK, 0, stream>>>(cums, particles, rnd, out);
}